// Test_P2B_iou_72954314490246
// MI455X (gfx1250) — compile-verified
//
#include <hip/hip_runtime.h>
#include <stdint.h>

// ---------------------------------------------------------------------------
// Bandwidth-bound IoU max + histogram + mean for MI455X (gfx1250).
// 256 MB of proposal boxes streamed once via global_load_async_to_lds_b128
// (ASYNCcnt-tracked DMA into LDS), double-buffered tiles, wave32 shfl-xor
// max reduction, LDS histogram, deterministic two-pass mean.
// ---------------------------------------------------------------------------

#define PROP            64
#define THREADS         256
#define WAVES           (THREADS / 32)
#define TILE_GTS        16
#define TILE_VEC        (TILE_GTS * PROP)          // float4 per tile = 1024 (16 KB)
#define LOADS_PER_THR   (TILE_VEC / THREADS)       // 4 async b128 per thread per tile
#define TILES_PER_BLOCK 8
#define GTS_PER_BLOCK   (TILE_GTS * TILES_PER_BLOCK)  // 128
#define GTS_PER_WAVE    (TILE_GTS / WAVES)            // 2 gts per wave per tile
#define NUM_BINS        50

__global__ __launch_bounds__(THREADS) void iou_hist_kernel(
    const float4* __restrict__ boxes,   // [N*P] (x1,y1,x2,y2)
    const float4* __restrict__ gts,     // [N]
    int*   __restrict__ out_hist,       // d_out[0..49]
    float* __restrict__ block_sums,     // d_ws, one float per block
    int n)
{
    __shared__ float4 s_boxes[2][TILE_VEC];   // 32 KB double buffer
    __shared__ int    s_hist[NUM_BINS];
    __shared__ float  s_wsum[WAVES];

    const int tid  = threadIdx.x;
    const int lane = tid & 31;
    const int wave = tid >> 5;

    if (tid < NUM_BINS) s_hist[tid] = 0;

    const long long blockGtBase = (long long)blockIdx.x * GTS_PER_BLOCK;
    const long long totalVec    = (long long)n * PROP;
    const uint64_t  baseAddr    = (uint64_t)(uintptr_t)boxes;

    // Issue one tile's worth of async global->LDS b128 copies (4 per thread).
    auto issue_tile = [&](int tile, int buf) {
        const long long vecBase = (blockGtBase + (long long)tile * TILE_GTS) * PROP;
        #pragma unroll
        for (int k = 0; k < LOADS_PER_THR; ++k) {
            const int t = tid + k * THREADS;
            const long long v = vecBase + t;
            // Clamp OOB lanes to offset 0 (compute side is guarded by gt<n).
            const uint32_t goff = (v < totalVec) ? (uint32_t)(v * 16) : 0u;
            // Generic LDS pointer: low 32 bits == LDS byte offset (ISA 10.2).
            const uint32_t loff = (uint32_t)(uintptr_t)(const void*)&s_boxes[buf][t];
            asm volatile("global_load_async_to_lds_b128 %0, %1, %2"
                         :
                         : "v"(loff), "v"(goff), "s"(baseAddr)
                         : "memory");
        }
    };

    issue_tile(0, 0);

    float lane_sum = 0.0f;   // only lane 0's copy is consumed

    for (int tile = 0; tile < TILES_PER_BLOCK; ++tile) {
        const int buf = tile & 1;
        if (tile + 1 < TILES_PER_BLOCK) {
            issue_tile(tile + 1, buf ^ 1);
            // Async loads complete in order: <=4 outstanding means the
            // current tile's 4 loads have landed in LDS.
            asm volatile("s_wait_asynccnt 4" ::: "memory");
        } else {
            asm volatile("s_wait_asynccnt 0" ::: "memory");
        }
        __syncthreads();   // tile visible to all waves (also covers s_hist init)

        #pragma unroll
        for (int j = 0; j < GTS_PER_WAVE; ++j) {
            const int localGt = wave * GTS_PER_WAVE + j;
            const long long gt = blockGtBase + (long long)tile * TILE_GTS + localGt;
            if (gt < n) {
                const float4 g = gts[gt];
                const float garea = (g.z - g.x) * (g.w - g.y);
                float m = 0.0f;   // IoU >= 0 always
                #pragma unroll
                for (int h = 0; h < 2; ++h) {
                    const float4 b = s_boxes[buf][localGt * PROP + lane + h * 32];
                    const float area1 = (b.z - b.x) * (b.w - b.y);
                    const float ltx = fmaxf(b.x, g.x);
                    const float lty = fmaxf(b.y, g.y);
                    const float rbx = fminf(b.z, g.z);
                    const float rby = fminf(b.w, g.w);
                    const float wx = fmaxf(rbx - ltx, 0.0f);
                    const float wy = fmaxf(rby - lty, 0.0f);
                    const float ov = wx * wy;
                    const float un = fmaxf(area1 + garea - ov, 1e-6f);
                    m = fmaxf(m, ov / un);
                }
                // wave32 butterfly max
                #pragma unroll
                for (int off = 16; off > 0; off >>= 1)
                    m = fmaxf(m, __shfl_xor(m, off, 32));
                if (lane == 0) {
                    int bin = (int)(m / 0.02f);       // matches floor(iou/0.02)
                    bin = bin > (NUM_BINS - 1) ? (NUM_BINS - 1) : bin;
                    atomicAdd(&s_hist[bin], 1);
                    lane_sum += m;
                }
            }
        }
        __syncthreads();   // all waves done with buf before it is refilled
    }

    if (lane == 0) s_wsum[wave] = lane_sum;
    __syncthreads();
    if (tid == 0) {
        float bs = 0.0f;
        #pragma unroll
        for (int w = 0; w < WAVES; ++w) bs += s_wsum[w];
        block_sums[blockIdx.x] = bs;     // deterministic: no float atomics
    }
    if (tid < NUM_BINS) {
        const int c = s_hist[tid];
        if (c) atomicAdd(&out_hist[tid], c);   // integer atomics: exact
    }
}

// Fixed-order reduction of per-block sums -> mean (deterministic).
__global__ __launch_bounds__(256) void finalize_kernel(
    const float* __restrict__ partials, int nparts,
    float* __restrict__ out_mean, float inv_n)
{
    __shared__ float red[256];
    float s = 0.0f;
    for (int i = threadIdx.x; i < nparts; i += 256) s += partials[i];
    red[threadIdx.x] = s;
    __syncthreads();
    for (int step = 128; step > 0; step >>= 1) {
        if (threadIdx.x < step) red[threadIdx.x] += red[threadIdx.x + step];
        __syncthreads();
    }
    if (threadIdx.x == 0) *out_mean = red[0] * inv_n;
}

extern "C" void kernel_launch(void* const* d_in, const int* in_sizes, int n_in,
                              void* d_out, int out_size, void* d_ws, size_t ws_size,
                              hipStream_t stream) {
    (void)n_in; (void)out_size; (void)ws_size;
    const float4* boxes = (const float4*)d_in[0];   // [N, P, 4] f32
    const float4* gts   = (const float4*)d_in[1];   // [N, 4]    f32
    const int n = in_sizes[1] / 4;                  // N

    int*   out_hist = (int*)d_out;                  // slots 0..49: int32 counts
    float* out_mean = (float*)d_out + NUM_BINS;     // slot 50: float mean
    float* block_sums = (float*)d_ws;

    const int nblocks = (n + GTS_PER_BLOCK - 1) / GTS_PER_BLOCK;

    hipMemsetAsync(d_out, 0, (NUM_BINS + 1) * sizeof(int), stream);

    iou_hist_kernel<<<nblocks, THREADS, 0, stream>>>(boxes, gts, out_hist,
                                                     block_sums, n);
    finalize_kernel<<<1, 256, 0, stream>>>(block_sums, nblocks, out_mean,
                                           1.0f / (float)n);
}